// Model_20598663151740
// MI455X (gfx1250) — compile-verified
//
#include <hip/hip_runtime.h>

// Fused upsample2x_bilinear + affine_grid + grid_sample_bilinear for
// x:(8,32,256,256) f32, theta:(8,2,3) f32 -> out:(8,32,512,512) f32.
//
// grid_sample(upsample2x(x)) collapses to a separable 4x4 stencil on x:
//   out = sum_{k=0..3} RW[k] * sum_{m=0..3} CW[m] * x[RI[k], CI[m]]
// with indices/weights shared across all 32 channels (grid depends only on
// n,yo,xo) and validity folded into the weights (separable: valid = vx*vy).
//
// MI455X strategy:
//  - One pass: 64MB read + 256MB write ~= 320MB HBM => ~14us floor @23.3TB/s.
//  - x (64MB) fits in the 192MB L2; gathers hit L2. Output stores use
//    non-temporal hint (TH=NT) so the 256MB write stream does not evict x.
//  - Per channel: issue all 16 gathers first (batched LOADcnt), then FMA.
//  - theta loads are scalar (readfirstlane on uniform batch index).

#define HN   8
#define HC   32
#define HIN  256
#define WIN  256
#define HOUT 512
#define WOUT 512

static __device__ __forceinline__ int imin(int a, int b) { return a < b ? a : b; }
static __device__ __forceinline__ int imax(int a, int b) { return a > b ? a : b; }

__global__ __launch_bounds__(256) void fused_affine_sample_kernel(
    const float* __restrict__ x,      // (8,32,256,256)
    const float* __restrict__ theta,  // (8,2,3)
    float* __restrict__ out)          // (8,32,512,512)
{
    const int id = blockIdx.x * 256 + threadIdx.x;   // position in [0, N*HOUT*WOUT)
    const int xo = id & (WOUT - 1);
    const int yo = (id >> 9) & (HOUT - 1);
    int n = id >> 18;
    // 262144 positions per batch => n uniform across a wave32; force SGPR so
    // the six theta loads become scalar s_load (KMcnt path, off the VMEM pipe).
    n = __builtin_amdgcn_readfirstlane(n);

    const float* t = theta + n * 6;
    const float gx = (float)(2 * xo + 1) * (1.0f / (float)WOUT) - 1.0f;
    const float gy = (float)(2 * yo + 1) * (1.0f / (float)HOUT) - 1.0f;
    const float sx = t[0] * gx + t[1] * gy + t[2];
    const float sy = t[3] * gx + t[4] * gy + t[5];

    // grid_sample source coords in upsampled (512x512) space
    const float ix = ((sx + 1.0f) * (float)WOUT - 1.0f) * 0.5f;
    const float iy = ((sy + 1.0f) * (float)HOUT - 1.0f) * 0.5f;

    const float ix0f = floorf(ix);
    const float iy0f = floorf(iy);
    const float wx = ix - ix0f;
    const float wy = iy - iy0f;
    const int ix0 = (int)ix0f;
    const int iy0 = (int)iy0f;

    int   CI[4]; float CW[4];   // column (x) source indices/weights into x (0..255)
    int   RI[4]; float RW[4];   // row (y) source indices/weights into x (0..255)

    #pragma unroll
    for (int b = 0; b < 2; ++b) {
        const int X = ix0 + b;
        const float wb  = b ? wx : (1.0f - wx);
        const float eff = (X >= 0 && X < WOUT) ? wb : 0.0f;  // separable validity
        const int Xc = imin(imax(X, 0), WOUT - 1);
        // upsample2x source: s = max((Xc+0.5)/2 - 0.5, 0); i0=floor(s); i1=min(i0+1,255)
        const float s = fmaxf(((float)Xc + 0.5f) * 0.5f - 0.5f, 0.0f);
        const int i0 = (int)s;                 // s >= 0 => trunc == floor
        const int i1 = imin(i0 + 1, WIN - 1);
        const float w = s - (float)i0;
        CI[2 * b]     = i0; CW[2 * b]     = eff * (1.0f - w);
        CI[2 * b + 1] = i1; CW[2 * b + 1] = eff * w;
    }
    #pragma unroll
    for (int a = 0; a < 2; ++a) {
        const int Y = iy0 + a;
        const float wa  = a ? wy : (1.0f - wy);
        const float eff = (Y >= 0 && Y < HOUT) ? wa : 0.0f;
        const int Yc = imin(imax(Y, 0), HOUT - 1);
        const float s = fmaxf(((float)Yc + 0.5f) * 0.5f - 0.5f, 0.0f);
        const int i0 = (int)s;
        const int i1 = imin(i0 + 1, HIN - 1);
        const float w = s - (float)i0;
        RI[2 * a]     = i0; RW[2 * a]     = eff * (1.0f - w);
        RI[2 * a + 1] = i1; RW[2 * a + 1] = eff * w;
    }

    // 16 tap offsets + combined weights, computed once, reused for all 32 channels.
    int   off[16];
    float wgt[16];
    #pragma unroll
    for (int k = 0; k < 4; ++k) {
        #pragma unroll
        for (int m = 0; m < 4; ++m) {
            off[k * 4 + m] = RI[k] * WIN + CI[m];
            wgt[k * 4 + m] = RW[k] * CW[m];
        }
    }

    const float* xn = x + (size_t)n * HC * (HIN * WIN);
    float* on = out + (((size_t)n * HC) * HOUT + yo) * (size_t)WOUT + xo;

    // Warm the cache line of the first tap (gfx1250 global_prefetch_b8).
    __builtin_prefetch(xn + off[0], 0, 0);

    #pragma unroll 8
    for (int c = 0; c < HC; ++c) {
        const float* __restrict__ p = xn + c * (HIN * WIN);
        // Phase 1: issue all 16 gathers (independent -> batched in LOADcnt,
        // backend can use staged s_wait_loadcnt instead of per-tap waits).
        float v[16];
        #pragma unroll
        for (int i = 0; i < 16; ++i) {
            v[i] = p[off[i]];
        }
        // Phase 2: weighted reduction (two partial chains to shorten the
        // dependency chain, then combine).
        float acc0 = 0.0f, acc1 = 0.0f;
        #pragma unroll
        for (int i = 0; i < 16; i += 2) {
            acc0 = fmaf(wgt[i],     v[i],     acc0);
            acc1 = fmaf(wgt[i + 1], v[i + 1], acc1);
        }
        // Non-temporal store: output (256MB) is write-once; keep it from
        // evicting x (64MB) out of the 192MB L2.
        __builtin_nontemporal_store(acc0 + acc1, on + (size_t)c * (HOUT * WOUT));
    }
}

extern "C" void kernel_launch(void* const* d_in, const int* in_sizes, int n_in,
                              void* d_out, int out_size, void* d_ws, size_t ws_size,
                              hipStream_t stream) {
    (void)in_sizes; (void)n_in; (void)out_size; (void)d_ws; (void)ws_size;
    const float* x     = (const float*)d_in[0];
    const float* theta = (const float*)d_in[1];
    float*       out   = (float*)d_out;

    const int positions = HN * HOUT * WOUT;      // 2,097,152
    const int block = 256;                       // 8 waves (wave32)
    const int grid  = positions / block;         // 8192 blocks
    fused_affine_sample_kernel<<<grid, block, 0, stream>>>(x, theta, out);
}